// ExpertPool_30502857736700
// MI455X (gfx1250) — compile-verified
//
#include <hip/hip_runtime.h>

typedef __attribute__((ext_vector_type(2))) float v2f;
typedef __attribute__((ext_vector_type(8))) float v8f;

#define M_BINS   10
#define N_BINS   8
#define NUM_E    80
#define D_IN     23
#define KP       24          // K padded to multiple of 4
#define H1       64
#define H2       32
#define B_TOTAL  65536
#define TILE_M   16
#define MAX_TILES (B_TOTAL / TILE_M + NUM_E)   // 4176 worst case

// -------- workspace layout (int32 offsets into d_ws) --------
#define WS_TILECOUNT 0
#define WS_COUNTS    16                        // 80 ints
#define WS_OFFSETS   112                       // 81 ints
#define WS_CURSORS   208                       // 80 ints
#define WS_TILEEXP   304                       // MAX_TILES ints
#define WS_TILESTART (WS_TILEEXP + MAX_TILES)  // MAX_TILES ints
#define WS_SORTED    (WS_TILESTART + MAX_TILES)// B_TOTAL ints

__global__ void init_kernel(int* ws) {
    int t = threadIdx.x;
    if (t < NUM_E) { ws[WS_COUNTS + t] = 0; ws[WS_CURSORS + t] = 0; }
}

__global__ void hist_kernel(const int* __restrict__ mon,
                            const int* __restrict__ mat, int* ws) {
    int i = blockIdx.x * blockDim.x + threadIdx.x;
    if (i < B_TOTAL) {
        int e = mon[i] * N_BINS + mat[i];
        atomicAdd(&ws[WS_COUNTS + e], 1);
    }
}

// 80 experts, <=4176 tiles: serial scan on one lane is cheap vs. the gather pass
__global__ void scan_kernel(int* ws) {
    if (threadIdx.x != 0) return;
    int off = 0;
    for (int e = 0; e < NUM_E; ++e) { ws[WS_OFFSETS + e] = off; off += ws[WS_COUNTS + e]; }
    ws[WS_OFFSETS + NUM_E] = off;
    int nt = 0;
    for (int e = 0; e < NUM_E; ++e) {
        int c = ws[WS_COUNTS + e], s = ws[WS_OFFSETS + e];
        for (int t = 0; t < c; t += TILE_M) {
            ws[WS_TILEEXP + nt]   = e;
            ws[WS_TILESTART + nt] = s + t;
            ++nt;
        }
    }
    ws[WS_TILECOUNT] = nt;
}

__global__ void scatter_kernel(const int* __restrict__ mon,
                               const int* __restrict__ mat, int* ws) {
    int i = blockIdx.x * blockDim.x + threadIdx.x;
    if (i < B_TOTAL) {
        int e = mon[i] * N_BINS + mat[i];
        int p = atomicAdd(&ws[WS_CURSORS + e], 1);
        ws[WS_SORTED + ws[WS_OFFSETS + e] + p] = i;
    }
}

// One wave32 per 16-sample tile; layers 1-2 via V_WMMA_F32_16X16X4_F32, layer 3 on VALU.
__global__ void __launch_bounds__(32)
moe_mlp_kernel(const float* __restrict__ x,
               const float* __restrict__ W1, const float* __restrict__ b1,
               const float* __restrict__ W2, const float* __restrict__ b2,
               const float* __restrict__ W3, const float* __restrict__ b3,
               const int* __restrict__ ws, float* __restrict__ out)
{
    __shared__ float xs [TILE_M * KP];   // 16x24 staged input rows (zero padded)
    __shared__ float h1s[TILE_M * H1];   // 16x64
    __shared__ float h2s[TILE_M * H2];   // 16x32

    const int tile = blockIdx.x;
    if (tile >= ws[WS_TILECOUNT]) return;     // uniform per wave

    const int lane  = threadIdx.x;
    const int half  = lane >> 4;       // 0: lanes 0-15, 1: lanes 16-31
    const int l15   = lane & 15;
    const int e     = ws[WS_TILEEXP + tile];
    const int start = ws[WS_TILESTART + tile];
    int nvalid = ws[WS_OFFSETS + e + 1] - start;
    if (nvalid > TILE_M) nvalid = TILE_M;

    // ---- stage gathered x rows into LDS; full wave: lane pairs split each row ----
    {
        const int r     = l15;
        const int jbase = half ? 12 : 0;
        const int jend  = half ? 24 : 12;          // upper half also writes pad col 23
        if (r < nvalid) {
            const int s = ws[WS_SORTED + start + r];
            const float* xr = x + (long)s * D_IN;
            #pragma unroll
            for (int j = 0; j < 12; ++j) {
                const int jj = jbase + j;
                xs[r * KP + jj] = (jj < D_IN) ? xr[jj] : 0.0f;   // only jj==23 pads
            }
        } else {
            for (int j = jbase; j < jend; ++j) xs[r * KP + j] = 0.0f;
        }
    }
    __syncthreads();

    // ---- layer 1: (16x24) * (24x64) ----
    v8f acc[4] = {};
    const float* W1e = W1 + e * (D_IN * H1);
    #pragma unroll
    for (int k0 = 0; k0 < 20; k0 += 4) {           // all K indices < 23: unguarded
        const int ka = k0 + half * 2;
        const v2f a = *(const v2f*)(xs + l15 * KP + ka);   // ds_load_b64
        #pragma unroll
        for (int nt = 0; nt < 4; ++nt) {
            const int col = nt * 16 + l15;
            v2f b;
            b.x = W1e[ ka      * H1 + col];
            b.y = W1e[(ka + 1) * H1 + col];
            acc[nt] = __builtin_amdgcn_wmma_f32_16x16x4_f32(
                false, a, false, b, (short)0, acc[nt], false, false);
        }
    }
    {   // peeled tail k0=20: k=23 must contribute zero (only b.y on upper half)
        const int ka = 20 + half * 2;              // 20 (lo) / 22 (hi)
        const v2f a = *(const v2f*)(xs + l15 * KP + ka);   // xs col 23 is zero anyway
        const int ky = half ? 22 : 21;             // clamp 23 -> valid row, value masked
        #pragma unroll
        for (int nt = 0; nt < 4; ++nt) {
            const int col = nt * 16 + l15;
            v2f b;
            b.x = W1e[ka * H1 + col];
            const float wy = W1e[ky * H1 + col];
            b.y = half ? 0.0f : wy;                // v_cndmask, no branch
            acc[nt] = __builtin_amdgcn_wmma_f32_16x16x4_f32(
                false, a, false, b, (short)0, acc[nt], false, false);
        }
    }
    #pragma unroll
    for (int nt = 0; nt < 4; ++nt) {
        const int col = nt * 16 + l15;
        const float bias = b1[e * H1 + col];
        #pragma unroll
        for (int r = 0; r < 8; ++r) {
            const int m = r + half * 8;            // C/D row mapping
            float v = acc[nt][r] + bias;
            h1s[m * H1 + col] = v > 0.0f ? v : 0.0f;
        }
    }
    __syncthreads();

    // ---- layer 2: (16x64) * (64x32) ----
    v8f acc2[2] = {};
    const float* W2e = W2 + e * (H1 * H2);
    #pragma unroll
    for (int k0 = 0; k0 < H1; k0 += 4) {
        const int ka = k0 + half * 2;
        const v2f a = *(const v2f*)(h1s + l15 * H1 + ka);  // ds_load_b64
        #pragma unroll
        for (int nt = 0; nt < 2; ++nt) {
            const int col = nt * 16 + l15;
            v2f b;
            b.x = W2e[ ka      * H2 + col];
            b.y = W2e[(ka + 1) * H2 + col];
            acc2[nt] = __builtin_amdgcn_wmma_f32_16x16x4_f32(
                false, a, false, b, (short)0, acc2[nt], false, false);
        }
    }
    #pragma unroll
    for (int nt = 0; nt < 2; ++nt) {
        const int col = nt * 16 + l15;
        const float bias = b2[e * H2 + col];
        #pragma unroll
        for (int r = 0; r < 8; ++r) {
            const int m = r + half * 8;
            float v = acc2[nt][r] + bias;
            h2s[m * H2 + col] = v > 0.0f ? v : 0.0f;
        }
    }
    __syncthreads();

    // ---- layer 3: (16x32) * (32x1) — 32 FMAs per row on VALU ----
    if (lane < nvalid) {                           // lanes 0..15 own rows
        const float* W3e = W3 + e * H2;
        float sum = b3[e];
        #pragma unroll
        for (int k = 0; k < H2; ++k) sum += h2s[lane * H2 + k] * W3e[k];
        const int s = ws[WS_SORTED + start + lane];
        out[s] = sum;
    }
}

extern "C" void kernel_launch(void* const* d_in, const int* in_sizes, int n_in,
                              void* d_out, int out_size, void* d_ws, size_t ws_size,
                              hipStream_t stream) {
    const float* x   = (const float*)d_in[0];
    const int*   mon = (const int*)  d_in[1];
    const int*   mat = (const int*)  d_in[2];
    const float* W1  = (const float*)d_in[3];
    const float* b1  = (const float*)d_in[4];
    const float* W2  = (const float*)d_in[5];
    const float* b2  = (const float*)d_in[6];
    const float* W3  = (const float*)d_in[7];
    const float* b3  = (const float*)d_in[8];
    float* out = (float*)d_out;
    int*   ws  = (int*)d_ws;

    init_kernel   <<<1, 128, 0, stream>>>(ws);
    hist_kernel   <<<(B_TOTAL + 255) / 256, 256, 0, stream>>>(mon, mat, ws);
    scan_kernel   <<<1, 32, 0, stream>>>(ws);
    scatter_kernel<<<(B_TOTAL + 255) / 256, 256, 0, stream>>>(mon, mat, ws);
    moe_mlp_kernel<<<MAX_TILES, 32, 0, stream>>>(x, W1, b1, W2, b2, W3, b3, ws, out);
}